// GNN_17325898072290
// MI455X (gfx1250) — compile-verified
//
#include <hip/hip_runtime.h>

#define D_IN  256
#define D_H1  128
#define D_H2  64
#define NCLS  40

typedef float v2f __attribute__((ext_vector_type(2)));
typedef float v8f __attribute__((ext_vector_type(8)));

// ---------------- degree / normalization ----------------

__global__ void init_deg_kernel(float* __restrict__ deg, int n) {
    int i = blockIdx.x * blockDim.x + threadIdx.x;
    if (i < n) deg[i] = 1.0f;   // self-loop weight
}

__global__ void edge_deg_kernel(const int* __restrict__ dst,
                                const float* __restrict__ ew,
                                float* __restrict__ deg, int e) {
    int i = blockIdx.x * blockDim.x + threadIdx.x;
    if (i < e) atomicAdd(&deg[dst[i]], ew[i]);
}

__global__ void dinv_kernel(float* __restrict__ deg, int n) {
    int i = blockIdx.x * blockDim.x + threadIdx.x;
    if (i < n) {
        float d = deg[i];
        deg[i] = (d > 0.0f) ? rsqrtf(d) : 0.0f;   // in-place deg -> dinv
    }
}

__global__ void norm_kernel(const int* __restrict__ src,
                            const int* __restrict__ dst,
                            const float* __restrict__ ew,
                            const float* __restrict__ dinv,
                            float* __restrict__ norm, int e) {
    int i = blockIdx.x * blockDim.x + threadIdx.x;
    if (i < e) norm[i] = dinv[src[i]] * ew[i] * dinv[dst[i]];
}

// ---------------- fp32 WMMA GEMM: H = X @ W ----------------
// X: [Nrows, K] row-major, W: [K, Nout] row-major, H: [Nrows, Nout].
// Register-blocked: one wave computes a 16x64 strip (4 accumulators), so the
// A fragment is loaded once per k-step and reused across 4 v_wmma issues.
// grid.x = ceil(rowTiles / wavesPerBlock), grid.y = Nout/64, block = 128.
//
// A 16x4 f32 layout : lane<16 holds (M=lane, K=k..k+1), lane>=16 (M=lane-16, K=k+2..k+3)
// B 4x16  f32 layout: v0 = rows K=k (lanes 0-15) / K=k+2 (lanes 16-31), v1 = K=k+1 / K=k+3
// C/D 16x16 layout  : VGPR r -> (M = r + 8*(lane>=16), N = lane&15)
__global__ void gemm_wmma_f32(const float* __restrict__ X,
                              const float* __restrict__ W,
                              float* __restrict__ H,
                              int Nrows, int K, int Nout) {
    const int lane = threadIdx.x & 31;
    const int wid  = threadIdx.x >> 5;
    const int half = lane >> 4;       // 0 or 1
    const int mrow = lane & 15;

    const long rowTile = (long)blockIdx.x * (blockDim.x >> 5) + wid;
    const long tileM   = rowTile * 16;
    if (tileM >= Nrows) return;                 // uniform per-wave guard
    const int  colBase = blockIdx.y * 64;       // 4 column tiles of 16

    v8f acc0 = {}, acc1 = {}, acc2 = {}, acc3 = {};
    const float* xrow = X + (tileM + mrow) * (long)K;
    const float* wcol = W + colBase + mrow;
    #pragma unroll 4
    for (int k = 0; k < K; k += 4) {
        const int ka = k + half * 2;
        const v2f a = *(const v2f*)(xrow + ka);          // single b64 load
        const float* w0 = wcol + (long)ka * Nout;
        const float* w1 = wcol + (long)(ka + 1) * Nout;
        v2f b0, b1, b2, b3;
        b0.x = w0[0];  b0.y = w1[0];
        b1.x = w0[16]; b1.y = w1[16];
        b2.x = w0[32]; b2.y = w1[32];
        b3.x = w0[48]; b3.y = w1[48];
        acc0 = __builtin_amdgcn_wmma_f32_16x16x4_f32(false, a, false, b0, (short)0, acc0, false, false);
        acc1 = __builtin_amdgcn_wmma_f32_16x16x4_f32(false, a, false, b1, (short)0, acc1, false, false);
        acc2 = __builtin_amdgcn_wmma_f32_16x16x4_f32(false, a, false, b2, (short)0, acc2, false, false);
        acc3 = __builtin_amdgcn_wmma_f32_16x16x4_f32(false, a, false, b3, (short)0, acc3, false, false);
    }
    float* hp = H + (tileM + half * 8) * (long)Nout + colBase + mrow;
    #pragma unroll
    for (int r = 0; r < 8; ++r) {
        hp[(long)r * Nout +  0] = acc0[r];
        hp[(long)r * Nout + 16] = acc1[r];
        hp[(long)r * Nout + 32] = acc2[r];
        hp[(long)r * Nout + 48] = acc3[r];
    }
}

// ---------------- edge aggregation: a[dst] += norm * h[src] ----------------
// One wave per edge; each lane handles 4 contiguous features (coalesced).
template<int F>
__global__ void edge_scatter_kernel(const int* __restrict__ src,
                                    const int* __restrict__ dst,
                                    const float* __restrict__ norm,
                                    const float* __restrict__ h,
                                    float* __restrict__ a, int e) {
    const long gid  = (long)blockIdx.x * blockDim.x + threadIdx.x;
    const long edge = gid >> 5;
    const int  lane = (int)(gid & 31);
    if (edge >= e) return;
    const int f = lane * 4;
    if (f >= F) return;
    const int   s = src[edge];
    const int   d = dst[edge];
    const float w = norm[edge];
    const float4 hv = *(const float4*)(h + (long)s * F + f);
    float* ap = a + (long)d * F + f;
    atomicAdd(ap + 0, w * hv.x);
    atomicAdd(ap + 1, w * hv.y);
    atomicAdd(ap + 2, w * hv.z);
    atomicAdd(ap + 3, w * hv.w);
}

// ---------------- node combine: a = relu(a + dinv^2 * h + b) ----------------
template<int F, bool RELU>
__global__ void node_combine_kernel(float* __restrict__ a,
                                    const float* __restrict__ h,
                                    const float* __restrict__ dinv,
                                    const float* __restrict__ bias,
                                    int n) {
    const int q   = F / 4;
    const int gid = blockIdx.x * blockDim.x + threadIdx.x;
    if (gid >= n * q) return;
    const int i = gid / q;
    const int f = (gid - i * q) * 4;
    const float di = dinv[i];
    const float s  = di * di;
    float4       av = *(float4*)(a + (long)i * F + f);
    const float4 hv = *(const float4*)(h + (long)i * F + f);
    const float4 bv = *(const float4*)(bias + f);
    av.x += s * hv.x + bv.x;
    av.y += s * hv.y + bv.y;
    av.z += s * hv.z + bv.z;
    av.w += s * hv.w + bv.w;
    if (RELU) {
        av.x = fmaxf(av.x, 0.0f);
        av.y = fmaxf(av.y, 0.0f);
        av.z = fmaxf(av.z, 0.0f);
        av.w = fmaxf(av.w, 0.0f);
    }
    *(float4*)(a + (long)i * F + f) = av;
}

// ---------------- classifier: out = h @ Wc + bc ----------------
// One thread per node row; 40 accumulators in registers, Wc stays hot in cache.
__global__ void classifier_kernel(const float* __restrict__ h,
                                  const float* __restrict__ Wc,
                                  const float* __restrict__ bc,
                                  float* __restrict__ out, int n) {
    const int i = blockIdx.x * blockDim.x + threadIdx.x;
    if (i >= n) return;
    float acc[NCLS];
    #pragma unroll
    for (int c = 0; c < NCLS; ++c) acc[c] = bc[c];
    const float* hr = h + (long)i * D_H2;
    for (int f = 0; f < D_H2; ++f) {
        const float xv = hr[f];
        #pragma unroll
        for (int c = 0; c < NCLS; ++c) acc[c] += xv * Wc[f * NCLS + c];
    }
    float* op = out + (long)i * NCLS;
    #pragma unroll
    for (int c = 0; c < NCLS; ++c) op[c] = acc[c];
}

// ---------------- host launch ----------------

extern "C" void kernel_launch(void* const* d_in, const int* in_sizes, int n_in,
                              void* d_out, int out_size, void* d_ws, size_t ws_size,
                              hipStream_t stream) {
    const float* x  = (const float*)d_in[0];
    const int*   ei = (const int*)d_in[1];
    const float* ew = (const float*)d_in[2];
    const float* W1 = (const float*)d_in[3];
    const float* b1 = (const float*)d_in[4];
    const float* W2 = (const float*)d_in[5];
    const float* b2 = (const float*)d_in[6];
    const float* Wc = (const float*)d_in[7];
    const float* bc = (const float*)d_in[8];
    float* out = (float*)d_out;

    const int n = in_sizes[0] / D_IN;   // 100000 (divisible by 16)
    const int e = in_sizes[2];          // 1600000
    const int* src = ei;
    const int* dst = ei + e;

    // Workspace layout (with reuse): dinv | norm | h1(=h2) | a1(=a2)  ~109 MB
    char* ws = (char*)d_ws;
    size_t off = 0;
    auto carve = [&](size_t bytes) -> float* {
        float* p = (float*)(ws + off);
        off += (bytes + 255) & ~(size_t)255;
        return p;
    };
    float* dinv = carve((size_t)n * 4);
    float* norm = carve((size_t)e * 4);
    float* h1   = carve((size_t)n * D_H1 * 4);
    float* a1   = carve((size_t)n * D_H1 * 4);
    float* h2   = h1;   // h1 dead after layer-1 combine
    float* a2   = a1;   // a1 dead after gemm2 reads it

    const int B = 256;
    const int rowTiles = n / 16;                  // 6250
    const int rowBlocks = (rowTiles + 3) / 4;     // 4 waves per block

    // Normalization (shared by both layers)
    init_deg_kernel<<<(n + B - 1) / B, B, 0, stream>>>(dinv, n);
    edge_deg_kernel<<<(e + B - 1) / B, B, 0, stream>>>(dst, ew, dinv, e);
    dinv_kernel<<<(n + B - 1) / B, B, 0, stream>>>(dinv, n);
    norm_kernel<<<(e + B - 1) / B, B, 0, stream>>>(src, dst, ew, dinv, norm, e);

    // ---- Layer 1: h1 = x @ W1 ; a1 = relu(scatter + self-loop + b1) ----
    {
        dim3 grid(rowBlocks, D_H1 / 64);   // wave = 16x64 strip
        gemm_wmma_f32<<<grid, 128, 0, stream>>>(x, W1, h1, n, D_IN, D_H1);
    }
    hipMemsetAsync(a1, 0, (size_t)n * D_H1 * 4, stream);
    {
        long threads = (long)e * 32;
        int blocks = (int)((threads + B - 1) / B);
        edge_scatter_kernel<D_H1><<<blocks, B, 0, stream>>>(src, dst, norm, h1, a1, e);
    }
    node_combine_kernel<D_H1, true><<<(n * (D_H1 / 4) + B - 1) / B, B, 0, stream>>>(
        a1, h1, dinv, b1, n);

    // ---- Layer 2: h2 = a1 @ W2 ; a2 = relu(scatter + self-loop + b2) ----
    {
        dim3 grid(rowBlocks, D_H2 / 64);   // grid.y = 1
        gemm_wmma_f32<<<grid, 128, 0, stream>>>(a1, W2, h2, n, D_H1, D_H2);
    }
    hipMemsetAsync(a2, 0, (size_t)n * D_H2 * 4, stream);
    {
        long threads = (long)e * 32;
        int blocks = (int)((threads + B - 1) / B);
        edge_scatter_kernel<D_H2><<<blocks, B, 0, stream>>>(src, dst, norm, h2, a2, e);
    }
    node_combine_kernel<D_H2, true><<<(n * (D_H2 / 4) + B - 1) / B, B, 0, stream>>>(
        a2, h2, dinv, b2, n);

    // ---- Classifier: out = a2 @ Wc + bc ----
    classifier_kernel<<<(n + B - 1) / B, B, 0, stream>>>(a2, Wc, bc, out, n);
}